// GCN_60129542144596
// MI455X (gfx1250) — compile-verified
//
#include <hip/hip_runtime.h>
#include <hip/hip_bf16.h>
#include <math.h>

// GCN 2-layer forward for MI455X (gfx1250, wave32).
// Dense projections use V_WMMA_F32_16X16X4_F32 (full fp32 precision, WMMA path).
// Graph scatter is edge-parallel with f32 global atomics (L2-resident on 192MB L2).

typedef float v2f __attribute__((ext_vector_type(2)));
typedef float v8f __attribute__((ext_vector_type(8)));

#define NF 128   // input features
#define NH 64    // hidden
#define NC 40    // classes

// ---------------------------------------------------------------- fill
__global__ void fill_f32(float* __restrict__ p, long n, float v) {
  long i = (long)blockIdx.x * blockDim.x + threadIdx.x;
  long stride = (long)gridDim.x * blockDim.x;
  for (; i < n; i += stride) p[i] = v;
}

// ---------------------------------------------------------------- degree
__global__ void degree_kernel(const long long* __restrict__ col,
                              const float* __restrict__ ew,
                              float* __restrict__ deg, int E) {
  int e = blockIdx.x * blockDim.x + threadIdx.x;
  if (e < E) atomicAdd(&deg[(int)col[e]], ew[e]);
}

__global__ void rsqrt_kernel(float* __restrict__ deg, int N) {
  int i = blockIdx.x * blockDim.x + threadIdx.x;
  if (i < N) {
    float d = deg[i];                       // >= 1 (self-loop folded into init)
    deg[i] = d > 0.f ? rsqrtf(fmaxf(d, 1e-30f)) : 0.f;
  }
}

// ---------------------------------------------------------------- GEMM1: h1 = x @ W1  (N x 128 @ 128 x 64)
// Block: 256 threads = 8 waves; 32 rows/block; wave tile 16x16; K-loop step 4.
__global__ __launch_bounds__(256)
void gemm1_wmma(const float* __restrict__ x, const float* __restrict__ W1,
                float* __restrict__ h1, int N) {
  __shared__ float Ws[NF][NH + 4];    // 128 x 68 (pad: bank-conflict-free column reads)
  __shared__ float Xs[32][NF + 4];    // 32 x 132

  const int t = threadIdx.x;
  const int base = blockIdx.x * 32;

  for (int idx = t; idx < NF * NH; idx += 256)
    Ws[idx / NH][idx % NH] = W1[idx];

  for (int idx = t; idx < 32 * NF; idx += 256) {
    int r = idx / NF, c = idx % NF;
    int node = base + r; if (node >= N) node = N - 1;   // clamp; tail rows not stored
    Xs[r][c] = x[(long)node * NF + c];
  }
  __syncthreads();

  const int wave = t >> 5, lane = t & 31;
  const int m0 = (wave >> 2) * 16;     // 0 / 16
  const int n0 = (wave & 3) * 16;      // 0..48
  const int lm = lane & 15, lh = lane >> 4;

  v8f c = {};
  for (int k0 = 0; k0 < NF; k0 += 4) {
    const int ka = k0 + lh * 2;
    v2f a, b;
    a.x = Xs[m0 + lm][ka];
    a.y = Xs[m0 + lm][ka + 1];
    b.x = Ws[ka][n0 + lm];
    b.y = Ws[ka + 1][n0 + lm];
    c = __builtin_amdgcn_wmma_f32_16x16x4_f32(false, a, false, b, (short)0, c,
                                              false, false);
  }
  for (int v = 0; v < 8; ++v) {
    int node = base + m0 + v + lh * 8;
    if (node < N) h1[(long)node * NH + n0 + lm] = c[v];
  }
}

// ---------------------------------------------------------------- scatter layer 1
// One wave32 per message unit (E edges + N self-loops). Lane handles 2 cols (float2).
__global__ __launch_bounds__(256)
void scatter1(const long long* __restrict__ ei, const float* __restrict__ ew,
              const float* __restrict__ dinv, const float* __restrict__ h1,
              float* __restrict__ a1, int N, int E) {
  long tid  = (long)blockIdx.x * blockDim.x + threadIdx.x;
  long unit = tid >> 5;
  int  lane = (int)(tid & 31);
  long total = (long)E + N;
  if (unit >= total) return;

  int r, cidx; float w;
  if (unit < E) { r = (int)ei[unit]; cidx = (int)ei[(long)E + unit]; w = ew[unit]; }
  else          { r = cidx = (int)(unit - E); w = 1.0f; }

  float nrm = dinv[r] * w * dinv[cidx];
  const float2 hv = ((const float2*)(h1 + (long)r * NH))[lane];
  float* dst = a1 + (long)cidx * NH + lane * 2;
  atomicAdd(dst,     nrm * hv.x);
  atomicAdd(dst + 1, nrm * hv.y);
}

// ---------------------------------------------------------------- bias + relu (in place)
__global__ void bias_relu(float* __restrict__ a1, const float* __restrict__ b1, long n) {
  long i = (long)blockIdx.x * blockDim.x + threadIdx.x;
  if (i < n) {
    float v = a1[i] + b1[(int)(i & (NH - 1))];
    a1[i] = v > 0.f ? v : 0.f;
  }
}

// ---------------------------------------------------------------- GEMM2: h2 = relu(a1) @ W2  (N x 64 @ 64 x 40)
// Block: 192 threads = 6 waves = 2 row-tiles x 3 col-tiles; cols padded 40->48.
__global__ __launch_bounds__(192)
void gemm2_wmma(const float* __restrict__ h, const float* __restrict__ W2,
                float* __restrict__ h2, int N) {
  __shared__ float Ws[NH][48 + 4];    // 64 x 52
  __shared__ float Xs[32][NH + 4];    // 32 x 68

  const int t = threadIdx.x;
  const int base = blockIdx.x * 32;

  for (int idx = t; idx < NH * 48; idx += 192) {
    int k = idx / 48, n = idx % 48;
    Ws[k][n] = (n < NC) ? W2[k * NC + n] : 0.f;
  }
  for (int idx = t; idx < 32 * NH; idx += 192) {
    int r = idx / NH, cc = idx % NH;
    int node = base + r; if (node >= N) node = N - 1;
    Xs[r][cc] = h[(long)node * NH + cc];
  }
  __syncthreads();

  const int wave = t >> 5, lane = t & 31;
  const int m0 = (wave / 3) * 16;
  const int n0 = (wave % 3) * 16;
  const int lm = lane & 15, lh = lane >> 4;

  v8f c = {};
  for (int k0 = 0; k0 < NH; k0 += 4) {
    const int ka = k0 + lh * 2;
    v2f a, b;
    a.x = Xs[m0 + lm][ka];
    a.y = Xs[m0 + lm][ka + 1];
    b.x = Ws[ka][n0 + lm];
    b.y = Ws[ka + 1][n0 + lm];
    c = __builtin_amdgcn_wmma_f32_16x16x4_f32(false, a, false, b, (short)0, c,
                                              false, false);
  }
  for (int v = 0; v < 8; ++v) {
    int node = base + m0 + v + lh * 8;
    int nn = n0 + lm;
    if (node < N && nn < NC) h2[(long)node * NC + nn] = c[v];
  }
}

// ---------------------------------------------------------------- scatter layer 2 (into d_out)
__global__ __launch_bounds__(256)
void scatter2(const long long* __restrict__ ei, const float* __restrict__ ew,
              const float* __restrict__ dinv, const float* __restrict__ h2,
              float* __restrict__ out, int N, int E) {
  long tid  = (long)blockIdx.x * blockDim.x + threadIdx.x;
  long unit = tid >> 5;
  int  lane = (int)(tid & 31);
  long total = (long)E + N;
  if (unit >= total) return;

  int r, cidx; float w;
  if (unit < E) { r = (int)ei[unit]; cidx = (int)ei[(long)E + unit]; w = ew[unit]; }
  else          { r = cidx = (int)(unit - E); w = 1.0f; }

  float nrm = dinv[r] * w * dinv[cidx];
  const float* hr = h2 + (long)r * NC;
  float* dst = out + (long)cidx * NC;
  atomicAdd(&dst[lane], nrm * hr[lane]);          // cols 0..31
  int c2 = lane + 32;                             // cols 32..39
  if (c2 < NC) atomicAdd(&dst[c2], nrm * hr[c2]);
}

// ---------------------------------------------------------------- bias + log_softmax (in place on d_out)
__global__ __launch_bounds__(256)
void logsoftmax_kernel(float* __restrict__ out, const float* __restrict__ b2, int N) {
  const int wid = threadIdx.x >> 5, lane = threadIdx.x & 31;
  const int node = blockIdx.x * 8 + wid;
  if (node >= N) return;
  float* p = out + (long)node * NC;

  float v0 = p[lane] + b2[lane];
  float v1 = (lane < 8) ? p[32 + lane] + b2[32 + lane] : -INFINITY;

  float mx = fmaxf(v0, v1);
  for (int off = 16; off; off >>= 1) mx = fmaxf(mx, __shfl_xor(mx, off, 32));
  float s = expf(v0 - mx) + ((lane < 8) ? expf(v1 - mx) : 0.f);
  for (int off = 16; off; off >>= 1) s += __shfl_xor(s, off, 32);
  float ls = mx + logf(s);

  p[lane] = v0 - ls;
  if (lane < 8) p[32 + lane] = v1 - ls;
}

// ---------------------------------------------------------------- launch
extern "C" void kernel_launch(void* const* d_in, const int* in_sizes, int n_in,
                              void* d_out, int out_size, void* d_ws, size_t ws_size,
                              hipStream_t stream) {
  const float*     x  = (const float*)d_in[0];
  const long long* ei = (const long long*)d_in[1];   // [2, E] int64
  const float*     ew = (const float*)d_in[2];
  const float*     W1 = (const float*)d_in[3];
  const float*     b1 = (const float*)d_in[4];
  const float*     W2 = (const float*)d_in[5];
  const float*     b2 = (const float*)d_in[6];

  const int N = in_sizes[0] / NF;
  const int E = in_sizes[2];

  float* ws   = (float*)d_ws;
  float* deg  = ws;                                   // N (dinv in place)
  float* h1   = ws + (((long)N + 63) & ~63L);         // N*64
  float* a1   = h1 + (long)N * NH;                    // N*64
  float* h2   = a1 + (long)N * NH;                    // N*40
  float* out  = (float*)d_out;                        // N*40 accumulator -> final

  const long units = (long)E + N;

  fill_f32<<<2048, 256, 0, stream>>>(deg, N, 1.0f);                // self-loop weight
  fill_f32<<<4096, 256, 0, stream>>>(a1, (long)N * NH, 0.0f);
  fill_f32<<<4096, 256, 0, stream>>>(out, (long)N * NC, 0.0f);

  degree_kernel<<<(E + 255) / 256, 256, 0, stream>>>(ei + E, ew, deg, E);
  rsqrt_kernel<<<(N + 255) / 256, 256, 0, stream>>>(deg, N);

  gemm1_wmma<<<(N + 31) / 32, 256, 0, stream>>>(x, W1, h1, N);

  scatter1<<<(int)((units + 7) / 8), 256, 0, stream>>>(ei, ew, deg, h1, a1, N, E);

  bias_relu<<<(int)(((long)N * NH + 255) / 256), 256, 0, stream>>>(a1, b1, (long)N * NH);

  gemm2_wmma<<<(N + 31) / 32, 192, 0, stream>>>(a1, W2, h2, N);

  scatter2<<<(int)((units + 7) / 8), 256, 0, stream>>>(ei, ew, deg, h2, out, N, E);

  logsoftmax_kernel<<<(N + 7) / 8, 256, 0, stream>>>(out, b2, N);
}